// Generator_72069551227427
// MI455X (gfx1250) — compile-verified
//
#include <hip/hip_runtime.h>
#include <math.h>
#include <limits.h>

// Reference constants: B = 200_000 graphs (its runtime value exists only as a
// 1-element device array, so scratch is sized with the source constant).
#define B_MAX 200000
#define BLK 256
#define MAX_BLOCKS 4096

// ---------------------------------------------------------------------------
// CDNA5 async global->LDS staging (ASYNCcnt path, CDNA5 ISA).
// Guarded so the file still compiles (with a synchronous fallback) if the
// toolchain lacks the builtins.
// ---------------------------------------------------------------------------
#define AS1 __attribute__((address_space(1)))
#define AS3 __attribute__((address_space(3)))

#if __has_builtin(__builtin_amdgcn_global_load_async_to_lds_b32)
#define HAVE_ASYNC 1
#else
#define HAVE_ASYNC 0
#endif

__device__ __forceinline__ void stage_b32(const void* g, void* lds) {
#if HAVE_ASYNC
    // flat LDS address: low 32 bits are the LDS offset (aperture rules), so an
    // integer truncation is the correct generic -> addrspace(3) conversion.
    __builtin_amdgcn_global_load_async_to_lds_b32(
        (AS1 int*)(uintptr_t)g,
        (AS3 int*)(uint32_t)(uintptr_t)lds,
        /*offset=*/0, /*cpol=*/0);
#else
    *(int*)lds = *(const int*)g;   // synchronous fallback through a VGPR
#endif
}

#if HAVE_ASYNC
#if __has_builtin(__builtin_amdgcn_s_wait_asynccnt)
#define WAIT_ASYNC(n) do { __builtin_amdgcn_s_wait_asynccnt(n); \
                           asm volatile("" ::: "memory"); } while (0)
#else
#define WAIT_ASYNC(n) asm volatile("s_wait_asynccnt " #n ::: "memory")
#endif
#define LDS_REUSE_FENCE() asm volatile("s_wait_dscnt 0x0" ::: "memory")
#else
#define WAIT_ASYNC(n) do { } while (0)
#define LDS_REUSE_FENCE() do { } while (0)
#endif

// ---- order-preserving float <-> int mapping (for global_atomic_max_i32) ----
__device__ __forceinline__ int f_ord(float f) {
    int b = __float_as_int(f);
    return (b >= 0) ? b : (b ^ 0x7fffffff);
}
__device__ __forceinline__ float f_unord(int m) {
    int b = (m >= 0) ? m : (m ^ 0x7fffffff);
    return __int_as_float(b);
}

// ---------------- kernel 0: init per-segment accumulators ----------------
__global__ __launch_bounds__(BLK)
void seg_init(int* __restrict__ smax, float* __restrict__ cnt,
              float* __restrict__ hsum, float* __restrict__ esum,
              float* __restrict__ dsum, int B) {
    int i = blockIdx.x * BLK + threadIdx.x;
    if (i < B) {
        smax[i] = INT_MIN;
        cnt[i]  = 0.0f;
        hsum[i] = 0.0f;
        esum[i] = 0.0f;
        dsum[i] = 0.0f;
    }
}

// ------- kernel 1: per-segment max(h/5), sum(h), count --------------------
// Tiled + double-buffered async staging of (h, batch) into LDS.
__global__ __launch_bounds__(BLK)
void pass_max_sum(const float* __restrict__ h, const int* __restrict__ batch,
                  const float* __restrict__ aug, const float* __restrict__ deg,
                  int* __restrict__ smax, float* __restrict__ cnt,
                  float* __restrict__ hsum, int n, int ntiles) {
    __shared__ float sh_h[2][BLK];
    __shared__ int   sh_b[2][BLK];
    const int tid    = threadIdx.x;
    const int lane   = tid & 31;
    const int stride = gridDim.x;

    int tile = blockIdx.x;
    if (tile < ntiles) {                       // prologue: stage first tile
        int j0 = tile * BLK + tid; j0 = (j0 < n) ? j0 : (n - 1);
        stage_b32(&h[j0],     &sh_h[0][tid]);
        stage_b32(&batch[j0], &sh_b[0][tid]);
    }
    int buf = 0;
    for (; tile < ntiles; tile += stride) {
        int next = tile + stride;
        if (next < ntiles) {                   // stage next tile, keep 2 in flight
            int jn = next * BLK + tid; jn = (jn < n) ? jn : (n - 1);
            LDS_REUSE_FENCE();
            stage_b32(&h[jn],     &sh_h[buf ^ 1][tid]);
            stage_b32(&batch[jn], &sh_b[buf ^ 1][tid]);
            WAIT_ASYNC(2);                     // current tile's 2 ops retired
        } else {
            WAIT_ASYNC(0);
        }

        int  gi    = tile * BLK + tid;
        bool valid = (gi < n);
        int   seg = sh_b[buf][tid];
        float hv  = valid ? sh_h[buf][tid] : 0.0f;
        float x   = valid ? (hv * 0.2f) : -INFINITY;  // identity for max
        float c   = valid ? 1.0f : 0.0f;
        float hs  = hv;

        // warm the 192MB L2 with pass-2 streams (gfx1250 global_prefetch_b8)
        int jp = valid ? gi : (n - 1);
        __builtin_prefetch(&aug[jp], 0, 0);
        __builtin_prefetch(&deg[jp], 0, 0);

        // wave32 segmented inclusive scan (batch is sorted)
#pragma unroll
        for (int off = 1; off < 32; off <<= 1) {
            float xm = __shfl_up(x,  off);
            float cm = __shfl_up(c,  off);
            float hm = __shfl_up(hs, off);
            int   sm = __shfl_up(seg, off);
            if (lane >= off && sm == seg) { x = fmaxf(x, xm); c += cm; hs += hm; }
        }
        int segdn = __shfl_down(seg, 1);
        if ((lane == 31) || (segdn != seg)) {
            atomicMax(&smax[seg], f_ord(x));   // global_atomic_max_i32
            atomicAdd(&cnt[seg],  c);          // global_atomic_add_f32
            atomicAdd(&hsum[seg], hs);
        }
        buf ^= 1;
    }
}

// -- kernel 2: e = exp(h/5 - m), delta = |hsum - aug|/(deg+1); seg sums -----
// Tiled + double-buffered async staging of (h, batch, aug, deg).
__global__ __launch_bounds__(BLK)
void pass_exp_delta(const float* __restrict__ h, const float* __restrict__ aug,
                    const float* __restrict__ deg, const int* __restrict__ batch,
                    const int* __restrict__ smax, const float* __restrict__ hsum,
                    float* __restrict__ e_out,   // aliases d_out's L slot
                    float* __restrict__ esum, float* __restrict__ dsum,
                    int n, int ntiles) {
    __shared__ float sh_h[2][BLK];
    __shared__ int   sh_b[2][BLK];
    __shared__ float sh_a[2][BLK];
    __shared__ float sh_d[2][BLK];
    const int tid    = threadIdx.x;
    const int lane   = tid & 31;
    const int stride = gridDim.x;

    int tile = blockIdx.x;
    if (tile < ntiles) {
        int j0 = tile * BLK + tid; j0 = (j0 < n) ? j0 : (n - 1);
        stage_b32(&h[j0],     &sh_h[0][tid]);
        stage_b32(&batch[j0], &sh_b[0][tid]);
        stage_b32(&aug[j0],   &sh_a[0][tid]);
        stage_b32(&deg[j0],   &sh_d[0][tid]);
    }
    int buf = 0;
    for (; tile < ntiles; tile += stride) {
        int next = tile + stride;
        if (next < ntiles) {
            int jn = next * BLK + tid; jn = (jn < n) ? jn : (n - 1);
            LDS_REUSE_FENCE();
            stage_b32(&h[jn],     &sh_h[buf ^ 1][tid]);
            stage_b32(&batch[jn], &sh_b[buf ^ 1][tid]);
            stage_b32(&aug[jn],   &sh_a[buf ^ 1][tid]);
            stage_b32(&deg[jn],   &sh_d[buf ^ 1][tid]);
            WAIT_ASYNC(4);
        } else {
            WAIT_ASYNC(0);
        }

        int  gi    = tile * BLK + tid;
        bool valid = (gi < n);
        int   seg = sh_b[buf][tid];
        float m   = f_unord(smax[seg]);                // L2-resident gather
        float hv  = sh_h[buf][tid];
        float e   = valid ? expf(hv * 0.2f - m) : 0.0f;
        float dlt = valid ? (fabsf(hsum[seg] - sh_a[buf][tid]) /
                             (sh_d[buf][tid] + 1.0f))
                          : 0.0f;
        if (valid) e_out[gi] = e;                      // stash numerator

#pragma unroll
        for (int off = 1; off < 32; off <<= 1) {
            float em = __shfl_up(e,   off);
            float dm = __shfl_up(dlt, off);
            int   sm = __shfl_up(seg, off);
            if (lane >= off && sm == seg) { e += em; dlt += dm; }
        }
        int segdn = __shfl_down(seg, 1);
        if ((lane == 31) || (segdn != seg)) {
            atomicAdd(&esum[seg], e);
            atomicAdd(&dsum[seg], dlt);
        }
        buf ^= 1;
    }
}

// ---------------- kernel 3: finalize L and bool_L ----------------
__global__ __launch_bounds__(BLK)
void pass_final(const float* __restrict__ aug, const float* __restrict__ deg,
                const int* __restrict__ batch,
                const float* __restrict__ hsum, const float* __restrict__ esum,
                const float* __restrict__ dsum, const float* __restrict__ cnt,
                float* __restrict__ L, float* __restrict__ boolL, int n) {
    int gi = blockIdx.x * BLK + threadIdx.x;
    if (gi >= n) return;
    int seg = batch[gi];
    float e = L[gi];                         // numerator stashed by pass 2
    L[gi] = e / esum[seg];
    float avg = dsum[seg] / fmaxf(cnt[seg], 1.0f);
    float dlt = fabsf(hsum[seg] - aug[gi]) / (deg[gi] + 1.0f);
    boolL[gi] = (dlt >= avg) ? 1.0f : 0.0f;
}

extern "C" void kernel_launch(void* const* d_in, const int* in_sizes, int n_in,
                              void* d_out, int out_size, void* d_ws, size_t ws_size,
                              hipStream_t stream) {
    const float* h     = (const float*)d_in[0];
    const float* aug   = (const float*)d_in[1];
    const float* deg   = (const float*)d_in[2];
    const int*   batch = (const int*)d_in[3];
    const int n = in_sizes[0];
    const int B = B_MAX;

    float* L     = (float*)d_out;       // [n]
    float* boolL = L + n;               // [n]

    // scratch partition: 5 * B * 4 bytes = 4 MB
    char* ws = (char*)d_ws;
    int*   smax = (int*)ws;    ws += (size_t)B * sizeof(int);
    float* cnt  = (float*)ws;  ws += (size_t)B * sizeof(float);
    float* hsum = (float*)ws;  ws += (size_t)B * sizeof(float);
    float* esum = (float*)ws;  ws += (size_t)B * sizeof(float);
    float* dsum = (float*)ws;

    const int ntiles = (n + BLK - 1) / BLK;
    const int gtiles = (ntiles < MAX_BLOCKS) ? ntiles : MAX_BLOCKS;

    dim3 blk(BLK);
    seg_init      <<<(B + BLK - 1) / BLK, blk, 0, stream>>>(smax, cnt, hsum, esum,
                                                            dsum, B);
    pass_max_sum  <<<gtiles, blk, 0, stream>>>(h, batch, aug, deg, smax, cnt,
                                               hsum, n, ntiles);
    pass_exp_delta<<<gtiles, blk, 0, stream>>>(h, aug, deg, batch, smax, hsum,
                                               L, esum, dsum, n, ntiles);
    pass_final    <<<ntiles, blk, 0, stream>>>(aug, deg, batch, hsum, esum, dsum,
                                               cnt, L, boolL, n);
}